// LogWignerCrystalSlaterFixedCYJastrow_5420248728035
// MI455X (gfx1250) — compile-verified
//
#include <hip/hip_runtime.h>

typedef float v2f __attribute__((ext_vector_type(2)));
typedef float v8f __attribute__((ext_vector_type(8)));

#define LBOX 20.0f
#define NTOT 64
#define NE   32
#define PI_F 3.14159265358979f

__device__ __forceinline__ float pmodf(float x, float m) {
    float r = fmodf(x, m);
    return (r < 0.0f) ? r + m : r;
}

// One block per walker. 256 threads = 8 wave32s.
// Phase 1: Jastrow over 64x64 pairs.
// Phase 2/3: per-spin 32x32 Slater matrix (WMMA Gram + rank-1 shift tables),
//            LU in LDS, logdet = sum log|U_kk|.
__global__ __launch_bounds__(256) void lwcs_fused_kernel(
    const float* __restrict__ walkerRs,   // (B, 64, 3)
    const float* __restrict__ log_alpha,  // scalar
    float* __restrict__ out)              // (B,)
{
    const int b    = blockIdx.x;
    const int tid  = threadIdx.x;
    const int lane = tid & 31;
    const int wave = tid >> 5;

    __shared__ float rx[NTOT], ry[NTOT], rz[NTOT];
    __shared__ float red[256];
    __shared__ float ex[NE], ey[NE], ez[NE];
    __shared__ float cxA[NE], cyA[NE], czA[NE];
    __shared__ float R2[NE], C2[NE];
    __shared__ float sG[NE][NE];        // Gram r_e . c_c (from WMMA)
    __shared__ float sPhi[NE][NE + 1];  // +1 pad vs LDS bank conflicts
    __shared__ float sRS[NE][27];       // r_e . s
    __shared__ float sCS[NE][27];       // c_c . s
    __shared__ float sS2[27];           // |s|^2
    __shared__ float sOut;

    // ---- load & wrap coordinates into LDS ----
    const float* wr = walkerRs + (size_t)b * NTOT * 3;
    if (tid < NTOT) {
        rx[tid] = pmodf(wr[tid * 3 + 0], LBOX);
        ry[tid] = pmodf(wr[tid * 3 + 1], LBOX);
        rz[tid] = pmodf(wr[tid * 3 + 2], LBOX);
    }
    const float la    = log_alpha[0];
    float alpha       = __expf(la);
    alpha             = fminf(fmaxf(alpha, 55.0f / (LBOX * LBOX)), 300.0f / (LBOX * LBOX));
    __syncthreads();

    // ======================= Jastrow =======================
    {
        const float dens  = (float)NTOT / (LBOX * LBOX * LBOX);
        const float A     = rsqrtf(4.0f * PI_F * dens);
        const float Fsame = sqrtf(2.0f * A);
        const float Fdiff = sqrtf(A);
        const int   i     = tid >> 2;       // 0..63
        const int   si    = i >> 5;
        const float xi = rx[i], yi = ry[i], zi = rz[i];
        float acc = 0.0f;
        for (int j = (tid & 3); j < NTOT; j += 4) {
            if (j == i) continue;
            float dx = xi - rx[j], dy = yi - ry[j], dz = zi - rz[j];
            dx = pmodf(dx + 0.5f * LBOX, LBOX) - 0.5f * LBOX;
            dy = pmodf(dy + 0.5f * LBOX, LBOX) - 0.5f * LBOX;
            dz = pmodf(dz + 0.5f * LBOX, LBOX) - 0.5f * LBOX;
            float r  = sqrtf(dx * dx + dy * dy + dz * dz);
            float F  = ((j >> 5) == si) ? Fsame : Fdiff;
            float cy = A / r * (1.0f - __expf(-r / F));
            float x  = fminf(r * (2.0f / LBOX), 1.0f - 1e-5f);
            float dec = __expf(1.0f - 1.0f / (1.0f - x * x));
            acc += cy * dec;
        }
        red[tid] = acc;
        __syncthreads();
        if (tid < 128) red[tid] += red[tid + 128];
        __syncthreads();
        if (tid < 64)  red[tid] += red[tid + 64];
        __syncthreads();
        if (tid < 32) {
            float v = red[tid] + red[tid + 32];
            for (int off = 16; off > 0; off >>= 1) v += __shfl_down(v, off, 32);
            if (tid == 0) sOut = -0.5f * v;
        }
    }

    // ======================= Slater (both spins) =======================
    const float peakAx  = 1.0f + 2.0f * __expf(-alpha * LBOX * LBOX);
    const float invPeak = 1.0f / (peakAx * peakAx * peakAx);

    for (int spin = 0; spin < 2; ++spin) {
        __syncthreads();
        // ---- electron coords + Wigner-crystal centers (n=4 cubic lattice, a=L/4) ----
        if (tid < NE) {
            float x = rx[spin * NE + tid];
            float y = ry[spin * NE + tid];
            float z = rz[spin * NE + tid];
            ex[tid] = x; ey[tid] = y; ez[tid] = z;
            R2[tid] = x * x + y * y + z * z;
            const float alat = LBOX * 0.25f;
            const float off  = spin ? 0.5f * alat : 0.0f;
            float cx = (float)((tid >> 4) & 3) * alat + off;
            float cc = (float)((tid >> 2) & 3) * alat + off;
            float cz = (float)(tid & 3) * alat + off;
            cxA[tid] = cx; cyA[tid] = cc; czA[tid] = cz;
            C2[tid]  = cx * cx + cc * cc + cz * cz;
        }
        __syncthreads();

        // ---- Gram matrix G[e][c] = r_e . c_c via V_WMMA_F32_16X16X4_F32 (wave 0) ----
        if (wave == 0) {
            #pragma unroll
            for (int mt = 0; mt < 2; ++mt) {
                const int m = mt * 16 + (lane & 15);
                v2f a;
                if (lane < 16) { a.x = ex[m]; a.y = ey[m]; }   // K=0, K=1
                else           { a.x = ez[m]; a.y = 0.0f; }    // K=2, K=3(pad)
                #pragma unroll
                for (int nt = 0; nt < 2; ++nt) {
                    const int n = nt * 16 + (lane & 15);
                    v2f bf;
                    if (lane < 16) { bf.x = cxA[n]; bf.y = cyA[n]; } // K=0, K=1
                    else           { bf.x = czA[n]; bf.y = 0.0f; }   // K=2, K=3
                    v8f c = {};
                    c = __builtin_amdgcn_wmma_f32_16x16x4_f32(
                            false, a, false, bf, (short)0, c, false, false);
                    #pragma unroll
                    for (int v = 0; v < 8; ++v) {
                        const int row = (lane < 16) ? v : v + 8;
                        sG[mt * 16 + row][nt * 16 + (lane & 15)] = c[v];
                    }
                }
            }
        }
        // ---- rank-1 image-shift tables ----
        for (int idx = tid; idx < 27; idx += 256) {
            float sx = (float)(idx / 9 - 1) * LBOX;
            float sy = (float)((idx / 3) % 3 - 1) * LBOX;
            float sz = (float)(idx % 3 - 1) * LBOX;
            sS2[idx] = sx * sx + sy * sy + sz * sz;
        }
        for (int idx = tid; idx < NE * 27; idx += 256) {
            int e = idx / 27, s = idx % 27;
            float sx = (float)(s / 9 - 1) * LBOX;
            float sy = (float)((s / 3) % 3 - 1) * LBOX;
            float sz = (float)(s % 3 - 1) * LBOX;
            sRS[e][s] = ex[e] * sx + ey[e] * sy + ez[e] * sz;
            sCS[e][s] = cxA[e] * sx + cyA[e] * sy + czA[e] * sz;
        }
        __syncthreads();

        // ---- Phi[e][c] = sum_s exp(-a*(R2+C2+|s|^2-2G-2r.s+2c.s)) / peak : 4 entries/thread
        {
            const int e  = tid >> 3;
            const int c0 = (tid & 7) * 4;
            for (int cc = 0; cc < 4; ++cc) {
                const int c   = c0 + cc;
                const float base = R2[e] + C2[c] - 2.0f * sG[e][c];
                float acc = 0.0f;
                #pragma unroll
                for (int s = 0; s < 27; ++s) {
                    float r2 = base + sS2[s] - 2.0f * sRS[e][s] + 2.0f * sCS[c][s];
                    acc += __expf(-alpha * r2);
                }
                sPhi[e][c] = acc * invPeak;
            }
        }
        __syncthreads();

        // ---- LU (no pivoting) in LDS: rows striped across 8 waves, lanes over columns
        for (int k = 0; k < NE - 1; ++k) {
            const float p = sPhi[k][k];
            for (int i = k + 1; i < NE; ++i) {
                if ((i & 7) != wave) continue;
                const float l = sPhi[i][k] / p;
                if (lane > k) sPhi[i][lane] -= l * sPhi[k][lane];
            }
            __syncthreads();
        }

        // ---- logdet = sum log|U_kk| ----
        {
            float v = 0.0f;
            if (tid < NE) v = __logf(fabsf(sPhi[tid][tid]));
            if (wave == 0) {
                for (int off = 16; off > 0; off >>= 1) v += __shfl_down(v, off, 32);
                if (lane == 0) sOut += v;
            }
        }
    }

    __syncthreads();
    if (tid == 0) out[b] = sOut;
}

extern "C" void kernel_launch(void* const* d_in, const int* in_sizes, int n_in,
                              void* d_out, int out_size, void* d_ws, size_t ws_size,
                              hipStream_t stream) {
    (void)n_in; (void)d_ws; (void)ws_size; (void)out_size;
    const float* walkerRs  = (const float*)d_in[0];
    const float* log_alpha = (const float*)d_in[1];
    float* out = (float*)d_out;
    const int B = in_sizes[0] / (NTOT * 3);
    lwcs_fused_kernel<<<B, 256, 0, stream>>>(walkerRs, log_alpha, out);
}